// Qwen3VLMoeTextExpertsWrapper_32461362823838
// MI455X (gfx1250) — compile-verified
//
#include <hip/hip_runtime.h>

// ---------------- problem constants (match reference) ----------------
#define T_TOKENS 1024      // B*S
#define HID      2048
#define INTER    768
#define TWO_I    1536
#define NEXP     32
#define TOPK     4
#define CAP      1024      // per-expert token list capacity

#define KC       32        // K-rows staged per TDM chunk
#define TG       4         // token tiles (16 tokens each) processed per pass

typedef float v2f  __attribute__((ext_vector_type(2)));
typedef float v8f  __attribute__((ext_vector_type(8)));
typedef unsigned int u32x4 __attribute__((ext_vector_type(4)));
typedef int   i32x4 __attribute__((ext_vector_type(4)));
typedef int   i32x8 __attribute__((ext_vector_type(8)));

// ---- Tensor Data Mover: 2D tile (tile_rows x tile_cols f32) global->LDS ----
// D# per CDNA5 ISA 8.3/8.4: group0 = {count|flags, lds_addr, global_addr, type},
// group1 = dims/tile/strides. Huge tensor_dim disables OOB clipping (we never
// read past real allocations: tiles are fully in-bounds). 6-arg builtin form
// (clang-23 / therock headers): trailing groups zero-filled (2D tensor).
__device__ __forceinline__ void tdm_load_tile_2d(unsigned lds_off,
                                                 const void* gptr,
                                                 int tile_cols, int tile_rows,
                                                 long long row_stride_elems) {
    unsigned long long ga = (unsigned long long)(uintptr_t)gptr;
    u32x4 g0;
    g0.x = 1u;                                                 // count=1, user D#
    g0.y = lds_off;                                            // lds_addr (bytes)
    g0.z = (unsigned)ga;                                       // global_addr[31:0]
    g0.w = (unsigned)((ga >> 32) & 0x01ffffffu) | (2u << 30);  // [56:32] | type=2
    const unsigned td0 = 0x00100000u;                          // 1M elems: no clip
    const unsigned td1 = 0x00100000u;
    unsigned long long st0 = (unsigned long long)row_stride_elems;
    i32x8 g1;
    g1[0] = (int)(2u << 16);                                   // data_size = 4B
    g1[1] = (int)((td0 & 0xffffu) << 16);                      // tensor_dim0 lo
    g1[2] = (int)(((td0 >> 16) & 0xffffu) | ((td1 & 0xffffu) << 16));
    g1[3] = (int)(((td1 >> 16) & 0xffffu) | (((unsigned)tile_cols & 0xffffu) << 16));
    g1[4] = (int)((unsigned)tile_rows & 0xffffu);              // tile_dim1, dim2=0
    g1[5] = (int)(unsigned)(st0 & 0xffffffffu);                // dim0_stride lo
    g1[6] = (int)(unsigned)((st0 >> 32) & 0xffffu);            // dim0_stride hi
    g1[7] = 0;
    i32x4 z4 = {0, 0, 0, 0};
    i32x8 z8 = {0, 0, 0, 0, 0, 0, 0, 0};
    __builtin_amdgcn_tensor_load_to_lds(g0, g1, z4, z4, z8, 0);
}

// ---------------- kernel 0: zero the output (we accumulate atomically) ----
__global__ void moe_zero_out(float* __restrict__ out, int n) {
    int i = blockIdx.x * blockDim.x + threadIdx.x;
    if (i < n) out[i] = 0.0f;
}

// ---------------- kernel 1: routing -> per-expert compact token lists ----
__global__ void moe_route(const int* __restrict__ router_indices,   // [T, K]
                          const float* __restrict__ routing_weights,// [T, E]
                          int* __restrict__ counts,
                          int* __restrict__ offsets,
                          int* __restrict__ tok,
                          float* __restrict__ wgt) {
    int e = threadIdx.x;
    if (e < NEXP) {
        int cnt = 0;
        for (int t = 0; t < T_TOKENS; ++t) {
            bool member = false;
            #pragma unroll
            for (int k = 0; k < TOPK; ++k)
                member |= (router_indices[t * TOPK + k] == e);
            if (member) {
                tok[e * CAP + cnt] = t;
                wgt[e * CAP + cnt] = routing_weights[t * NEXP + e];
                ++cnt;
            }
        }
        counts[e] = cnt;
    }
    __syncthreads();
    if (threadIdx.x == 0) {
        int acc = 0;
        for (int i = 0; i < NEXP; ++i) { offsets[i] = acc; acc += counts[i]; }
        offsets[NEXP] = acc;
    }
}

// ---------------- kernel 2: gate_up GEMM + SwiGLU + combine weight -------
// grid.x = NEXP * (INTER/64). 256 threads = 8 waves.
// Weight tiles (KC x 64 gate + KC x 64 up) are TDM-streamed into double-
// buffered LDS; B operands come from LDS, A (gathered x rows) from global
// (L2-resident). 4 token tiles share each staged chunk (4 accumulators).
__global__ void __launch_bounds__(256)
moe_gateup(const float* __restrict__ x,        // [T, H]
           const float* __restrict__ gup,      // [E, H, 2I]
           const int*   __restrict__ counts,
           const int*   __restrict__ offsets,
           const int*   __restrict__ tok,
           const float* __restrict__ wgt,
           float*       __restrict__ act) {    // [pairs, I] compact
    const int NT = 64;
    int e    = blockIdx.x / (INTER / NT);
    int nt   = blockIdx.x % (INTER / NT);
    int wv   = threadIdx.x >> 5;
    int lane = threadIdx.x & 31;
    int cnt  = counts[e];
    if (cnt == 0) return;
    int off  = offsets[e];

    const float* W = gup + (size_t)e * HID * TWO_I;
    const int colbase_gate = nt * NT;
    const int colbase_up   = INTER + nt * NT;

    __shared__ float wbuf[2][2][KC * NT];   // [parity][gate/up][KC][64] = 32KB
    __shared__ float tile[8 * 16 * 16];     // C exchange, 8KB

    int region = wv >> 2;                        // 0 = gate, 1 = up
    int wcol   = (wv & 3) * 16 + (lane & 15);
    int koff   = (lane >> 4) * 2;                // 0 or 2

    const int nch = HID / KC;                    // 64 chunks
    int ntt = (cnt + 15) >> 4;

    for (int tt0 = 0; tt0 < ntt; tt0 += TG) {
        // A-row pointers for the 4 token tiles (padded lanes read row 0)
        const float* xrow[TG];
        #pragma unroll
        for (int g = 0; g < TG; ++g) {
            int slot = (tt0 + g) * 16 + (lane & 15);
            int t = (slot < cnt) ? tok[e * CAP + slot] : 0;
            xrow[g] = x + (size_t)t * HID + koff;
        }

        v8f c[TG];
        #pragma unroll
        for (int g = 0; g < TG; ++g)
            c[g] = (v8f){0.f, 0.f, 0.f, 0.f, 0.f, 0.f, 0.f, 0.f};

        // preload chunks 0 and 1 (wave 0 only; TDM issues once per wave)
        if (wv == 0) {
            #pragma unroll
            for (int ch = 0; ch < 2; ++ch) {
                tdm_load_tile_2d((unsigned)(uintptr_t)&wbuf[ch][0][0],
                                 W + (size_t)(ch * KC) * TWO_I + colbase_gate,
                                 NT, KC, TWO_I);
                tdm_load_tile_2d((unsigned)(uintptr_t)&wbuf[ch][1][0],
                                 W + (size_t)(ch * KC) * TWO_I + colbase_up,
                                 NT, KC, TWO_I);
            }
        }

        for (int ch = 0; ch < nch; ++ch) {
            // chunk ch complete when <=2 tensor ops (chunk ch+1) outstanding
            if (ch + 1 < nch) __builtin_amdgcn_s_wait_tensorcnt(2);
            else              __builtin_amdgcn_s_wait_tensorcnt(0);
            __syncthreads();

            int p = ch & 1;
            const float* wb = &wbuf[p][region][0];
            #pragma unroll
            for (int s = 0; s < KC / 4; ++s) {
                v2f b;
                b.x = wb[(4 * s + koff)     * NT + wcol];
                b.y = wb[(4 * s + koff + 1) * NT + wcol];
                #pragma unroll
                for (int g = 0; g < TG; ++g) {
                    v2f a = *(const v2f*)(xrow[g] + ch * KC + 4 * s);
                    c[g] = __builtin_amdgcn_wmma_f32_16x16x4_f32(
                               false, a, false, b, (short)0, c[g], false, false);
                }
            }
            __syncthreads();    // all reads of buffer p done
            if (wv == 0 && ch + 2 < nch) {
                tdm_load_tile_2d((unsigned)(uintptr_t)&wbuf[p][0][0],
                                 W + (size_t)((ch + 2) * KC) * TWO_I + colbase_gate,
                                 NT, KC, TWO_I);
                tdm_load_tile_2d((unsigned)(uintptr_t)&wbuf[p][1][0],
                                 W + (size_t)((ch + 2) * KC) * TWO_I + colbase_up,
                                 NT, KC, TWO_I);
            }
        }

        // epilogue: per token tile, exchange C via LDS, fused SwiGLU * cw
        #pragma unroll 1
        for (int g = 0; g < TG; ++g) {
            __syncthreads();
            #pragma unroll
            for (int r = 0; r < 8; ++r)
                tile[wv * 256 + (r + 8 * (lane >> 4)) * 16 + (lane & 15)] = c[g][r];
            __syncthreads();
            int tt = tt0 + g;
            for (int idx = threadIdx.x; idx < 16 * NT; idx += 256) {
                int m  = idx / NT;
                int cc = idx % NT;
                int s2 = tt * 16 + m;
                if (s2 < cnt) {
                    float gg = tile[(cc >> 4) * 256       + m * 16 + (cc & 15)];
                    float uu = tile[((cc >> 4) + 4) * 256 + m * 16 + (cc & 15)];
                    float silu = gg / (1.0f + __expf(-gg));
                    float cw = wgt[e * CAP + s2];
                    act[(size_t)(off + s2) * INTER + nt * NT + cc] = silu * uu * cw;
                }
            }
        }
    }
}

// ---------------- kernel 3: down proj GEMM + atomic scatter-accumulate ---
// grid.x = NEXP * (HID/128). 256 threads = 8 waves, each owns 16 H-columns.
// down-weight tiles (KC x 128) TDM-streamed into double-buffered LDS.
__global__ void __launch_bounds__(256)
moe_down(const float* __restrict__ act,        // [pairs, I]
         const float* __restrict__ dwn,        // [E, I, H]
         const int*   __restrict__ counts,
         const int*   __restrict__ offsets,
         const int*   __restrict__ tok,
         float*       __restrict__ out) {      // [T, H]
    const int HT = 128;
    int e    = blockIdx.x / (HID / HT);
    int ht   = blockIdx.x % (HID / HT);
    int wv   = threadIdx.x >> 5;
    int lane = threadIdx.x & 31;
    int cnt  = counts[e];
    if (cnt == 0) return;
    int off  = offsets[e];

    const float* W = dwn + (size_t)e * INTER * HID + ht * HT;

    __shared__ float wbuf[2][KC * HT];          // 2 x 32x128 f32 = 32KB

    int wcol = wv * 16 + (lane & 15);
    int koff = (lane >> 4) * 2;
    int col  = ht * HT + wcol;

    const int nch = INTER / KC;                 // 24 chunks
    int ntt = (cnt + 15) >> 4;

    for (int tt0 = 0; tt0 < ntt; tt0 += TG) {
        const float* arow[TG];
        #pragma unroll
        for (int g = 0; g < TG; ++g) {
            int slot = (tt0 + g) * 16 + (lane & 15);
            int slot_c = (slot < cnt) ? slot : (cnt - 1);
            arow[g] = act + (size_t)(off + slot_c) * INTER + koff;
        }

        v8f c[TG];
        #pragma unroll
        for (int g = 0; g < TG; ++g)
            c[g] = (v8f){0.f, 0.f, 0.f, 0.f, 0.f, 0.f, 0.f, 0.f};

        if (wv == 0) {
            #pragma unroll
            for (int ch = 0; ch < 2; ++ch)
                tdm_load_tile_2d((unsigned)(uintptr_t)&wbuf[ch][0],
                                 W + (size_t)(ch * KC) * HID, HT, KC, HID);
        }

        for (int ch = 0; ch < nch; ++ch) {
            if (ch + 1 < nch) __builtin_amdgcn_s_wait_tensorcnt(1);
            else              __builtin_amdgcn_s_wait_tensorcnt(0);
            __syncthreads();

            int p = ch & 1;
            const float* wb = &wbuf[p][0];
            #pragma unroll
            for (int s = 0; s < KC / 4; ++s) {
                v2f b;
                b.x = wb[(4 * s + koff)     * HT + wcol];
                b.y = wb[(4 * s + koff + 1) * HT + wcol];
                #pragma unroll
                for (int g = 0; g < TG; ++g) {
                    v2f a = *(const v2f*)(arow[g] + ch * KC + 4 * s);
                    c[g] = __builtin_amdgcn_wmma_f32_16x16x4_f32(
                               false, a, false, b, (short)0, c[g], false, false);
                }
            }
            __syncthreads();
            if (wv == 0 && ch + 2 < nch)
                tdm_load_tile_2d((unsigned)(uintptr_t)&wbuf[p][0],
                                 W + (size_t)((ch + 2) * KC) * HID, HT, KC, HID);
        }

        #pragma unroll
        for (int g = 0; g < TG; ++g) {
            #pragma unroll
            for (int r = 0; r < 8; ++r) {
                int m  = r + 8 * (lane >> 4);
                int s2 = (tt0 + g) * 16 + m;
                if (s2 < cnt) {
                    int t = tok[e * CAP + s2];
                    unsafeAtomicAdd(&out[(size_t)t * HID + col], c[g][r]);
                }
            }
        }
    }
}

// ---------------- host-side launcher ----------------
extern "C" void kernel_launch(void* const* d_in, const int* in_sizes, int n_in,
                              void* d_out, int out_size, void* d_ws, size_t ws_size,
                              hipStream_t stream) {
    const float* hidden_states   = (const float*)d_in[0];  // [B,S,H]
    const float* routing_weights = (const float*)d_in[1];  // [B,S,E]
    const float* gate_up_proj    = (const float*)d_in[2];  // [E,H,2I]
    const float* down_proj       = (const float*)d_in[3];  // [E,I,H]
    const int*   router_indices  = (const int*)d_in[4];    // [B,S,K]
    float* out = (float*)d_out;

    char* ws = (char*)d_ws;
    int*   counts  = (int*)ws;                          //  32 ints
    int*   offsets = counts + 32;                       //  33 ints
    int*   tok     = (int*)(ws + 512);                  //  32*1024 ints
    float* wgt     = (float*)(ws + 512 + NEXP*CAP*4);   //  32*1024 floats
    float* act     = (float*)(ws + 512 + 2*NEXP*CAP*4); //  4096*768 floats

    const int out_elems = T_TOKENS * HID;
    moe_zero_out<<<(out_elems + 1023) / 1024, 1024, 0, stream>>>(out, out_elems);

    moe_route<<<1, 32, 0, stream>>>(router_indices, routing_weights,
                                    counts, offsets, tok, wgt);

    moe_gateup<<<NEXP * (INTER / 64), 256, 0, stream>>>(
        hidden_states, gate_up_proj, counts, offsets, tok, wgt, act);

    moe_down<<<NEXP * (HID / 128), 256, 0, stream>>>(
        act, down_proj, counts, offsets, tok, out);
}